// MuxGNNLayer_10239202033919
// MI455X (gfx1250) — compile-verified
//
#include <hip/hip_runtime.h>

#define N_NODES 100000
#define R_REL   4
#define E_EDGES 1600000
#define DD      128
#define DIM_A   128
#define LN_EPS  1e-5f

typedef float v2f __attribute__((ext_vector_type(2)));
typedef float v8f __attribute__((ext_vector_type(8)));

// ---------------------------------------------------------------------------
// fp32 WMMA: D(16x16) = A(16x4) * B(4x16) + C   (V_WMMA_F32_16X16X4_F32)
// ---------------------------------------------------------------------------
__device__ __forceinline__ v8f wmma_f32_16x16x4(v2f a, v2f b, v8f c) {
    return __builtin_amdgcn_wmma_f32_16x16x4_f32(
        /*neg_a=*/false, a, /*neg_b=*/false, b,
        /*c_mod=*/(short)0, c, /*reuse_a=*/false, /*reuse_b=*/false);
}

__device__ __forceinline__ void atomic_add_f32(float* p, float v) {
    __hip_atomic_fetch_add(p, v, __ATOMIC_RELAXED, __HIP_MEMORY_SCOPE_AGENT);
}

// ---------------------------------------------------------------------------
// K0: rst[r][n][d] = node_feat[n][r][d]   (float4 copy / transpose)
// ---------------------------------------------------------------------------
__global__ void k_init_rst(const float* __restrict__ node_feat, float* __restrict__ rst) {
    long long idx = (long long)blockIdx.x * blockDim.x + threadIdx.x;
    const long long total = (long long)R_REL * N_NODES * (DD / 4);
    if (idx >= total) return;
    int d4 = (int)(idx % (DD / 4));
    long long t = idx / (DD / 4);
    int n = (int)(t % N_NODES);
    int r = (int)(t / N_NODES);
    const float4* s = (const float4*)node_feat;
    float4* d = (float4*)rst;
    d[((long long)r * N_NODES + n) * (DD / 4) + d4] =
        s[((long long)n * R_REL + r) * (DD / 4) + d4];
}

// ---------------------------------------------------------------------------
// K1: one wave per edge: rst[r][dst] += node_feat[src][r]  (f32 atomics, L2-resident)
// ---------------------------------------------------------------------------
__global__ void k_edge_scatter(const float* __restrict__ node_feat,
                               const int* __restrict__ src,
                               const int* __restrict__ dst,
                               float* __restrict__ rst) {
    int wave = threadIdx.x >> 5;
    int lane = threadIdx.x & 31;
    long long g = (long long)blockIdx.x * (blockDim.x >> 5) + wave;
    if (g >= (long long)R_REL * E_EDGES) return;
    int r = (int)(g / E_EDGES);
    int e = (int)(g % E_EDGES);
    int sn = src[(long long)r * E_EDGES + e];
    int dn = dst[(long long)r * E_EDGES + e];
    float4 v = ((const float4*)node_feat)[((long long)sn * R_REL + r) * (DD / 4) + lane];
    float* o = rst + ((long long)r * N_NODES + dn) * DD + lane * 4;
    atomic_add_f32(o + 0, v.x);
    atomic_add_f32(o + 1, v.y);
    atomic_add_f32(o + 2, v.z);
    atomic_add_f32(o + 3, v.w);
}

// ---------------------------------------------------------------------------
// K2/K3: in-place X = relu(X @ W + b), X is (R*N, 128), W is (128,128).
// Block owns 16 rows; 8 waves = 8 col-tiles of 16. __syncthreads separates the
// K-loop reads from the in-place writes (block's rows are exclusive to it).
// ---------------------------------------------------------------------------
__global__ void k_gemm_relu_inplace(float* __restrict__ X,
                                    const float* __restrict__ W,
                                    const float* __restrict__ bias) {
    int wave = threadIdx.x >> 5;
    int lane = threadIdx.x & 31;
    long long row0 = (long long)blockIdx.x * 16;
    int col0 = wave * 16;
    int half = lane >> 4;   // 0: lanes 0-15, 1: lanes 16-31
    int l15  = lane & 15;
    const float* arow = X + (row0 + l15) * DD;
    v8f acc = {};
    #pragma unroll 4
    for (int k0 = 0; k0 < DD; k0 += 4) {
        v2f a, b;
        // A 16x4 f32 layout: VGPR0/1 = K {k0,k0+1} (lanes 0-15) / {k0+2,k0+3} (lanes 16-31)
        a.x = arow[k0 + 2 * half];
        a.y = arow[k0 + 2 * half + 1];
        // B 4x16: row-striped across lanes; same K split across lane halves
        b.x = W[(k0 + 2 * half) * DD + col0 + l15];
        b.y = W[(k0 + 2 * half + 1) * DD + col0 + l15];
        acc = wmma_f32_16x16x4(a, b, acc);
    }
    __syncthreads();   // all reads of this block's 16 rows complete before in-place writes
    float bv = bias[col0 + l15];
    #pragma unroll
    for (int j = 0; j < 8; ++j) {
        int m = j + 8 * half;                 // C/D layout: VGPR j -> rows j / j+8
        float v = acc[j] + bv;
        X[(row0 + m) * DD + col0 + l15] = v > 0.f ? v : 0.f;
    }
}

// ---------------------------------------------------------------------------
// K4: LayerNorm per row of 128 (one wave per row, shuffle-tree reductions)
// ---------------------------------------------------------------------------
__global__ void k_layernorm(float* __restrict__ H,
                            const float* __restrict__ gamma,
                            const float* __restrict__ beta) {
    int wave = threadIdx.x >> 5;
    int lane = threadIdx.x & 31;
    long long row = (long long)blockIdx.x * (blockDim.x >> 5) + wave;
    if (row >= (long long)R_REL * N_NODES) return;
    float4* rowp = (float4*)(H + row * DD);
    float4 v = rowp[lane];
    float s = v.x + v.y + v.z + v.w;
    for (int m = 16; m >= 1; m >>= 1) s += __shfl_xor(s, m, 32);
    float mu = s * (1.0f / DD);
    float cx = v.x - mu, cy = v.y - mu, cz = v.z - mu, cw = v.w - mu;
    float q = cx * cx + cy * cy + cz * cz + cw * cw;
    for (int m = 16; m >= 1; m >>= 1) q += __shfl_xor(q, m, 32);
    float rinv = rsqrtf(q * (1.0f / DD) + LN_EPS);
    float4 g = ((const float4*)gamma)[lane];
    float4 b = ((const float4*)beta)[lane];
    float4 o;
    o.x = cx * rinv * g.x + b.x;
    o.y = cy * rinv * g.y + b.y;
    o.z = cz * rinv * g.z + b.z;
    o.w = cw * rinv * g.w + b.w;
    rowp[lane] = o;
}

// ---------------------------------------------------------------------------
// K5: scores[r][n][k] = (tanh(H[r] @ Ws1[r]) @ Ws2[r])[n][k]
// Block = (r, 16-row slab). Phase 1: 8 waves compute tanh tile into LDS (WMMA).
// Phase 2: 64 threads do 128-long dots against Ws2 (128x4).
// ---------------------------------------------------------------------------
__global__ void k_scores(const float* __restrict__ H,
                         const float* __restrict__ Ws1,
                         const float* __restrict__ Ws2,
                         float* __restrict__ scores) {
    __shared__ float t[16 * DIM_A];
    int wave = threadIdx.x >> 5;
    int lane = threadIdx.x & 31;
    int r  = blockIdx.x / (N_NODES / 16);
    int mt = blockIdx.x % (N_NODES / 16);
    long long row0 = (long long)r * N_NODES + (long long)mt * 16;
    const float* W = Ws1 + (long long)r * DD * DIM_A;
    int col0 = wave * 16;
    int half = lane >> 4;
    int l15  = lane & 15;
    const float* arow = H + (row0 + l15) * DD;
    v8f acc = {};
    #pragma unroll 4
    for (int k0 = 0; k0 < DD; k0 += 4) {
        v2f a, b;
        a.x = arow[k0 + 2 * half];
        a.y = arow[k0 + 2 * half + 1];
        b.x = W[(k0 + 2 * half) * DIM_A + col0 + l15];
        b.y = W[(k0 + 2 * half + 1) * DIM_A + col0 + l15];
        acc = wmma_f32_16x16x4(a, b, acc);
    }
    #pragma unroll
    for (int j = 0; j < 8; ++j) {
        int m = j + 8 * half;
        t[m * DIM_A + col0 + l15] = tanhf(acc[j]);
    }
    __syncthreads();
    if (threadIdx.x < 16 * R_REL) {
        int m = threadIdx.x >> 2;
        int k = threadIdx.x & 3;
        const float* w2 = Ws2 + (long long)r * DIM_A * R_REL;
        float s = 0.f;
        for (int d = 0; d < DIM_A; ++d) s += t[m * DIM_A + d] * w2[d * R_REL + k];
        long long n = (long long)mt * 16 + m;
        scores[((long long)r * N_NODES + n) * R_REL + k] = s;
    }
}

// ---------------------------------------------------------------------------
// K6: softmax over relations + out[n][r][:] = sum_k attn[n][r][k] * H[k][n][:]
// One wave per node; lane owns 4 of the 128 features.
// ---------------------------------------------------------------------------
__global__ void k_attn_out(const float* __restrict__ H,
                           const float* __restrict__ scores,
                           float* __restrict__ out) {
    int wave = threadIdx.x >> 5;
    int lane = threadIdx.x & 31;
    long long n = (long long)blockIdx.x * (blockDim.x >> 5) + wave;
    if (n >= N_NODES) return;
    float a[R_REL][R_REL];
    #pragma unroll
    for (int r = 0; r < R_REL; ++r)
        #pragma unroll
        for (int k = 0; k < R_REL; ++k)
            a[r][k] = scores[((long long)r * N_NODES + n) * R_REL + k];
    #pragma unroll
    for (int k = 0; k < R_REL; ++k) {
        float mx = a[0][k];
        #pragma unroll
        for (int r = 1; r < R_REL; ++r) mx = fmaxf(mx, a[r][k]);
        float sum = 0.f;
        #pragma unroll
        for (int r = 0; r < R_REL; ++r) { a[r][k] = __expf(a[r][k] - mx); sum += a[r][k]; }
        float inv = 1.f / sum;
        #pragma unroll
        for (int r = 0; r < R_REL; ++r) a[r][k] *= inv;
    }
    float4 o[R_REL];
    #pragma unroll
    for (int r = 0; r < R_REL; ++r) o[r] = make_float4(0.f, 0.f, 0.f, 0.f);
    #pragma unroll
    for (int k = 0; k < R_REL; ++k) {
        float4 hv = ((const float4*)(H + ((long long)k * N_NODES + n) * DD))[lane];
        #pragma unroll
        for (int r = 0; r < R_REL; ++r) {
            o[r].x += a[r][k] * hv.x;
            o[r].y += a[r][k] * hv.y;
            o[r].z += a[r][k] * hv.z;
            o[r].w += a[r][k] * hv.w;
        }
    }
    #pragma unroll
    for (int r = 0; r < R_REL; ++r)
        ((float4*)(out + ((long long)n * R_REL + r) * DD))[lane] = o[r];
}

// ---------------------------------------------------------------------------
extern "C" void kernel_launch(void* const* d_in, const int* in_sizes, int n_in,
                              void* d_out, int out_size, void* d_ws, size_t ws_size,
                              hipStream_t stream) {
    const float* node_feat = (const float*)d_in[0];
    const int*   src       = (const int*)d_in[1];
    const int*   dst       = (const int*)d_in[2];
    const float* W1        = (const float*)d_in[3];
    const float* b1        = (const float*)d_in[4];
    const float* W2        = (const float*)d_in[5];
    const float* b2        = (const float*)d_in[6];
    const float* Ws1       = (const float*)d_in[7];
    const float* Ws2       = (const float*)d_in[8];
    const float* gamma     = (const float*)d_in[9];
    const float* beta      = (const float*)d_in[10];

    float* H      = (float*)d_ws;                                  // (R*N, 128): rst -> h1 -> h (in-place)
    float* scores = H + (size_t)R_REL * N_NODES * DD;              // (R, N, R)
    float* out    = (float*)d_out;

    {   // K0: rst = node_feat (transposed)
        long long tot = (long long)R_REL * N_NODES * (DD / 4);
        k_init_rst<<<(int)((tot + 255) / 256), 256, 0, stream>>>(node_feat, H);
    }
    {   // K1: edge scatter-add
        long long waves = (long long)R_REL * E_EDGES;
        k_edge_scatter<<<(int)((waves + 7) / 8), 256, 0, stream>>>(node_feat, src, dst, H);
    }
    // K2/K3: two fused GEMM+bias+ReLU passes, in-place
    k_gemm_relu_inplace<<<(R_REL * N_NODES) / 16, 256, 0, stream>>>(H, W1, b1);
    k_gemm_relu_inplace<<<(R_REL * N_NODES) / 16, 256, 0, stream>>>(H, W2, b2);
    {   // K4: LayerNorm
        long long waves = (long long)R_REL * N_NODES;
        k_layernorm<<<(int)((waves + 7) / 8), 256, 0, stream>>>(H, gamma, beta);
    }
    // K5: attention scores
    k_scores<<<R_REL * (N_NODES / 16), 256, 0, stream>>>(H, Ws1, Ws2, scores);
    // K6: softmax over relations + output mix
    k_attn_out<<<(N_NODES + 7) / 8, 256, 0, stream>>>(H, scores, out);
}